// Group_11802570130410
// MI455X (gfx1250) — compile-verified
//
#include <hip/hip_runtime.h>
#include <hip/hip_bf16.h>
#include <float.h>

// Problem constants from the reference setup.
#define BB 4
#define NN 16384
#define MM 4096
#define CC 64
#define KK 32

typedef __attribute__((ext_vector_type(2))) float v2f;
typedef __attribute__((ext_vector_type(8))) float v8f;

// ---------------------------------------------------------------------------
// Kernel 0: point squared norms xx[b][n] = sum_d x[b][d][n]^2   (B*N floats)
// ---------------------------------------------------------------------------
__global__ __launch_bounds__(256) void knn_norms(const float* __restrict__ X,
                                                 float* __restrict__ XX) {
    int i = blockIdx.x * blockDim.x + threadIdx.x;  // over B*N
    if (i >= BB * NN) return;
    int b = i / NN, n = i % NN;
    const float* Xb = X + (size_t)b * 3 * NN;
    float x0 = Xb[n], x1 = Xb[NN + n], x2 = Xb[2 * NN + n];
    XX[i] = x0 * x0 + x1 * x1 + x2 * x2;
}

// WGP(CU)-scope prefetch: pulls into ALL cache levels (the builtin only
// reaches SE/SYS scope, which stops at GL2). Speculative: OOB is dropped.
__device__ __forceinline__ void prefetch_wgp(const void* p) {
    asm volatile("global_prefetch_b8 %0, off" :: "v"(p));
}

// ---------------------------------------------------------------------------
// Kernel A: distances via V_WMMA_F32_16X16X4_F32 + per-lane top-K.
//
// One wave (32 lanes) owns 16 queries. B operand (queries, K=4):
//   lanes 0-15 : { -2*y0, -2*y1 }   (K=0, K=1)
//   lanes 16-31: { -2*y2,  1.0  }   (K=2, K=3)
// A operand per 16-point tile:
//   lanes 0-15 : {  x0,   x1  }
//   lanes 16-31: {  x2,  xx   }
// C operand = yy broadcast  =>  D = xx + yy - 2*dot = squared distance.
// C/D layout: lane holds query q = lane&15; VGPR v holds point row
//   p = n0 + v + (lane>=16 ? 8 : 0).
//
// 4 tiles per loop iteration: 8 clause-grouped loads in flight, 4
// independent WMMAs with staged loadcnt waits, WGP-scope prefetch two
// 64-point chunks ahead. Candidate lists are (dist,idx) float2 entries in
// LDS so merge scans fetch both halves with one b64 op.
// ---------------------------------------------------------------------------
__global__ __launch_bounds__(128) void knn_topk(const float* __restrict__ X,
                                                const float* __restrict__ Y,
                                                const float* __restrict__ XX,
                                                int* __restrict__ IDX) {
    __shared__ float2 s_cand[4][KK][32];   // [wave][slot][lane] = (dist, idx)

    const int wave = threadIdx.x >> 5;
    const int lane = threadIdx.x & 31;
    const int q    = lane & 15;
    const int half = lane >> 4;

    const int tilesPerBatch = MM / 64;                 // 64 queries per block
    const int b  = blockIdx.x / tilesPerBatch;
    const int mt = blockIdx.x % tilesPerBatch;
    const int m  = mt * 64 + wave * 16 + q;

    const float* Xb  = X  + (size_t)b * 3 * NN;
    const float* Yb  = Y  + (size_t)b * 3 * MM;
    const float* XXb = XX + (size_t)b * NN;

    // --- loop-invariant B and C operands ---
    const float y0 = Yb[m], y1 = Yb[MM + m], y2 = Yb[2 * MM + m];
    const float yy = y0 * y0 + y1 * y1 + y2 * y2;
    v2f bop;
    if (half == 0) { bop.x = -2.0f * y0; bop.y = -2.0f * y1; }
    else           { bop.x = -2.0f * y2; bop.y = 1.0f;       }
    v8f cop;
#pragma unroll
    for (int v = 0; v < 8; ++v) cop[v] = yy;

    // --- init per-lane candidate list (LDS column owned by this lane) ---
#pragma unroll
    for (int s = 0; s < KK; ++s)
        s_cand[wave][s][lane] = make_float2(FLT_MAX, __int_as_float(0));
    float cur_max  = FLT_MAX;
    int   max_slot = 0;

    // --- per-lane A-operand source pointers (half-dependent) ---
    const int pa = lane & 15;
    const float* a0p = (half == 0) ? (Xb + pa)      : (Xb + 2 * NN + pa);
    const float* a1p = (half == 0) ? (Xb + NN + pa) : (XXb + pa);

    for (int n0 = 0; n0 < NN; n0 += 64) {
        // WGP-scope prefetch two chunks ahead so the next clause of loads
        // hits in the WGP cache (tail overrun dropped harmlessly).
        prefetch_wgp(a0p + n0 + 128);
        prefetch_wgp(a1p + n0 + 128);

        // 8 loads issued before any consumer -> overlapped traffic.
        v2f a[4];
#pragma unroll
        for (int t = 0; t < 4; ++t) {
            a[t].x = a0p[n0 + 16 * t];
            a[t].y = a1p[n0 + 16 * t];
        }

        // 4 independent WMMAs.
        v8f d[4];
#pragma unroll
        for (int t = 0; t < 4; ++t)
            d[t] = __builtin_amdgcn_wmma_f32_16x16x4_f32(
                false, a[t], false, bop, (short)0, cop, false, false);

        // Candidate insertion (replace-max, LDS-backed list).
#pragma unroll
        for (int t = 0; t < 4; ++t) {
            const int pbase = n0 + 16 * t + half * 8;
#pragma unroll
            for (int v = 0; v < 8; ++v) {
                float dv = fmaxf(d[t][v], 0.0f);  // clamp; also maps NaN -> 0
                if (dv < cur_max) {
                    s_cand[wave][max_slot][lane] =
                        make_float2(dv, __int_as_float(pbase + v));
                    float mx = -FLT_MAX; int ms = 0;
#pragma unroll
                    for (int s = 0; s < KK; ++s) {
                        float tval = s_cand[wave][s][lane].x;  // b32 read
                        if (tval > mx) { mx = tval; ms = s; }
                    }
                    cur_max = mx; max_slot = ms;
                }
            }
        }
    }

    __syncthreads();

    // --- merge the two half-lists per query & emit ascending order ---
    if (half == 0) {
        const int out_base = (b * MM + m) * KK;
        for (int k = 0; k < KK; ++k) {
            float best = FLT_MAX; int bidx = 0x7fffffff;
            int bs = 0, bc = lane;
#pragma unroll 1
            for (int s = 0; s < KK; ++s) {
#pragma unroll
                for (int c = 0; c < 2; ++c) {
                    int col = lane + c * 16;
                    float2 e = s_cand[wave][s][col];        // one b64 load
                    float t  = e.x;
                    int   ti = __float_as_int(e.y);
                    if (t < best || (t == best && ti < bidx)) {
                        best = t; bidx = ti; bs = s; bc = col;
                    }
                }
            }
            IDX[out_base + k] = bidx;
            s_cand[wave][bs][bc].x = FLT_MAX;   // remove selected entry
        }
    }
}

// ---------------------------------------------------------------------------
// Kernel B: gather. One wave per (b, m); lane = k.
// out[b][c][m][k] = (c<3) ? x[b][c][idx] - y[b][c][m] : f[b][c-3][idx]
// Output is write-once (140 MB) -> non-temporal stores keep the gather
// working set (X/F, ~17.6 MB) resident in the 192 MB L2.
// ---------------------------------------------------------------------------
__global__ __launch_bounds__(256) void knn_gather(const float* __restrict__ X,
                                                  const float* __restrict__ Y,
                                                  const float* __restrict__ F,
                                                  const int* __restrict__ IDX,
                                                  float* __restrict__ OUT) {
    const int gw   = (blockIdx.x * blockDim.x + threadIdx.x) >> 5;  // over B*M
    const int lane = threadIdx.x & 31;
    if (gw >= BB * MM) return;
    const int b = gw / MM, m = gw % MM;

    const int id = IDX[((size_t)b * MM + m) * KK + lane];

    const float* Xb = X + (size_t)b * 3 * NN;
    const float* Yb = Y + (size_t)b * 3 * MM;
    const float* Fb = F + (size_t)b * CC * NN;
    float* Ob = OUT + (size_t)b * (3 + CC) * MM * KK;
    const size_t mk = (size_t)m * KK + lane;

#pragma unroll
    for (int d = 0; d < 3; ++d) {
        float v = Xb[(size_t)d * NN + id] - Yb[(size_t)d * MM + m];
        __builtin_nontemporal_store(v, &Ob[(size_t)d * MM * KK + mk]);
    }
#pragma unroll 8
    for (int c = 0; c < CC; ++c) {
        float v = Fb[(size_t)c * NN + id];
        __builtin_nontemporal_store(v, &Ob[(size_t)(3 + c) * MM * KK + mk]);
    }
}

// ---------------------------------------------------------------------------
extern "C" void kernel_launch(void* const* d_in, const int* in_sizes, int n_in,
                              void* d_out, int out_size, void* d_ws, size_t ws_size,
                              hipStream_t stream) {
    const float* X = (const float*)d_in[0];   // points       [B,3,N]
    const float* Y = (const float*)d_in[1];   // new_points   [B,3,M]
    const float* F = (const float*)d_in[2];   // features     [B,C,N]
    // d_in[3] = num_samples (hard-coded KK=32)
    float* OUT = (float*)d_out;               // [B, 3+C, M, K]

    float* XX  = (float*)d_ws;                                   // B*N floats
    int*   IDX = (int*)((char*)d_ws + (size_t)BB * NN * sizeof(float)); // B*M*K ints

    knn_norms <<<(BB * NN + 255) / 256, 256, 0, stream>>>(X, XX);
    knn_topk  <<<BB * (MM / 64),        128, 0, stream>>>(X, Y, XX, IDX);
    knn_gather<<<(BB * MM) / 8,         256, 0, stream>>>(X, Y, F, IDX, OUT);
}